// HH_Synaptic_71012989272304
// MI455X (gfx1250) — compile-verified
//
#include <hip/hip_runtime.h>
#include <stdint.h>

// ---------------------------------------------------------------------------
// Hodgkin-Huxley + synapse scan for MI455X (gfx1250).
//   z   : (B=32, N=2000, L=128) float32
//   out : (B=32, N=2000, L=128) float32  = sigmoid((V - VT)/KP)
// One thread per neuron (b,l): 4096 threads = 128 wave32 blocks.
// z is streamed Global->LDS with the Tensor Data Mover (TDM), double-buffered,
// synchronized with s_wait_tensorcnt.  Inner step is pure registers with
// native v_exp_f32 / v_rcp_f32.
// ---------------------------------------------------------------------------

typedef __attribute__((ext_vector_type(4))) unsigned int uint32x4_t;
typedef __attribute__((ext_vector_type(8))) int          int32x8_t;
typedef __attribute__((ext_vector_type(4))) int          int32x4_t;

#define B_      32
#define N_      2000
#define L_      128
#define LCHUNK  32                          // lanes per block (wave32)
#define TCHUNK  128                         // timesteps per TDM tile
#define NZ      (N_ - 1)                    // 1999 z-steps actually consumed
#define NCHUNK  ((NZ + TCHUNK - 1) / TCHUNK)

__device__ __forceinline__ float fexp(float x) {
  // exp(x) = exp2(x * log2(e)) -> v_exp_f32
  return __builtin_amdgcn_exp2f(x * 1.44269504088896340736f);
}

__device__ __forceinline__ float frcp(float x) {
  // v_rcp_f32 + one Newton-Raphson refinement
  float r = __builtin_amdgcn_rcpf(x);
  return r * (2.0f - x * r);
}

// Issue one TDM load: a (TCHUNK x LCHUNK) float tile from global (row stride
// L_ floats) into LDS.  tensor_dim1 = `remaining` rows so out-of-range rows of
// the final partial tile read as zero instead of touching memory past z.
__device__ __forceinline__ void tdm_load_tile(const float* gptr, float* lptr,
                                              int remaining) {
  uint64_t ga = (uint64_t)(uintptr_t)gptr;
  uint32_t la = (uint32_t)(uintptr_t)lptr;  // low 32 bits of flat = LDS offset

  // ---- D# group 0 (128b): count=1 | lds_addr | global_addr | type=2 -------
  uint32x4_t g0;
  g0.x = 1u;                                   // count=1, is_restore=0
  g0.y = la;                                   // lds_addr (bytes)
  g0.z = (uint32_t)ga;                         // global_addr[31:0]
  g0.w = (uint32_t)(ga >> 32) | (2u << 30);    // global_addr[56:32] | type=2

  // ---- D# group 1 (256b), packed via 64-bit words ------------------------
  // [17:16]  data_size=2 (4B)       [79:48]   tensor_dim0 = 32
  // [111:80] tensor_dim1 = remaining[127:112] tile_dim0   = 32
  // [143:128] tile_dim1 = TCHUNK    [207:160] tensor_dim0_stride = 128
  // everything else (masks, padding, barrier, dim1_stride) = 0
  uint64_t td0  = (uint64_t)LCHUNK;            // tensor_dim0
  uint64_t td1  = (uint64_t)(uint32_t)(remaining > 0 ? remaining : 0);
  uint64_t s0   = (uint64_t)L_;                // tensor_dim0_stride (elements)
  uint64_t q0 = (2ull << 16)                   // data_size = 4 bytes
              | ((td0 & 0xFFFFull) << 48);     // tensor_dim0[15:0]
  uint64_t q1 = (td0 >> 16)                    // tensor_dim0[31:16]
              | (td1 << 16)                    // tensor_dim1
              | ((uint64_t)LCHUNK << 48);      // tile_dim0
  uint64_t q2 = (uint64_t)TCHUNK               // tile_dim1
              | ((s0 & 0xFFFFFFFFull) << 32);  // tensor_dim0_stride[31:0]
  uint64_t q3 = (s0 >> 32);                    // tensor_dim0_stride[47:32]

  int32x8_t g1;
  g1[0] = (int)(uint32_t)q0;  g1[1] = (int)(uint32_t)(q0 >> 32);
  g1[2] = (int)(uint32_t)q1;  g1[3] = (int)(uint32_t)(q1 >> 32);
  g1[4] = (int)(uint32_t)q2;  g1[5] = (int)(uint32_t)(q2 >> 32);
  g1[6] = (int)(uint32_t)q3;  g1[7] = (int)(uint32_t)(q3 >> 32);

  int32x4_t gz4  = {0, 0, 0, 0};               // groups 2/3 unused (2D tensor)
  int32x8_t gz8  = {0, 0, 0, 0, 0, 0, 0, 0};   // extra group (clang-23 form)

  // clang-23 / amdgpu-toolchain 6-arg form
  __builtin_amdgcn_tensor_load_to_lds(g0, g1, gz4, gz4, gz8, 0);
}

__device__ __forceinline__ float cn_update(float a, float b, float x) {
  // Crank-Nicolson gating update: (a*DT + (1-s)*x) / (1+s), s = DT/2*(a+b)
  float s = 0.025f * (a + b);
  return (a * 0.05f + (1.0f - s) * x) * frcp(s + 1.0f);
}

__global__ void __launch_bounds__(LCHUNK)
hh_synaptic_kernel(const float* __restrict__ z, float* __restrict__ out) {
  const int lane = threadIdx.x;              // 0..31
  const int b    = blockIdx.x >> 2;          // batch
  const int lc   = blockIdx.x & 3;           // which 32-wide slice of L

  __shared__ float zbuf[2][TCHUNK * LCHUNK]; // 2 x 16 KB double buffer

  const float* zbase = z   + (size_t)b * N_ * L_ + (size_t)lc * LCHUNK;
  float*       obase = out + (size_t)b * N_ * L_ + (size_t)lc * LCHUNK;

  // ---- kick off first tile --------------------------------------------------
  tdm_load_tile(zbase, &zbuf[0][0], NZ);

  // ---- state (registers) ----------------------------------------------------
  float V = -70.0f, m = 0.0f, n = 0.0f, h = 1.0f, y = 0.0f;

  // t = 0 output: sigmoid((V0 - VT)/KP), VT=-20, KP=3
  obase[lane] = frcp(1.0f + fexp(-(V + 20.0f) * (1.0f / 3.0f)));

  int t = 1;
  for (int c = 0; c < NCHUNK; ++c) {
    const int base  = c * TCHUNK;
    const int steps = (NZ - base < TCHUNK) ? (NZ - base) : TCHUNK;

    if (c + 1 < NCHUNK) {
      // prefetch next tile into the other buffer, then wait for current tile
      tdm_load_tile(zbase + (size_t)(base + TCHUNK) * L_,
                    &zbuf[(c + 1) & 1][0], NZ - (base + TCHUNK));
      __builtin_amdgcn_s_wait_tensorcnt(1);
    } else {
      __builtin_amdgcn_s_wait_tensorcnt(0);
    }

    const float* zb = &zbuf[c & 1][0];
    for (int s = 0; s < steps; ++s) {
      const float zp = zb[s * LCHUNK + lane];   // ds_load_b32, conflict-free

      // --- membrane voltage (semi-implicit) ---
      float m3 = m * m * m;
      float n2 = n * n;
      float p1 = 40.0f * m3 * h;                // GNA * m^3 * h
      float p2 = 35.0f * n2 * n2;               // GK  * n^4
      float G  = 0.025f * (p1 + p2 + 0.1f + 0.5f * y);
      float E  = p1 * 55.0f + p2 * (-77.0f) + 0.1f * (-65.0f); // GS*VS*y == 0
      float Vn = (V * (1.0f - G) + 0.05f * (E + 0.5f)) * frcp(1.0f + G);

      // --- rate constants (6 native exps) ---
      float aN = 0.02f  * (Vn - 25.0f) * frcp(1.0f - fexp((25.0f - Vn) * (1.0f / 9.0f)));
      float aM = 0.182f * (Vn + 35.0f) * frcp(1.0f - fexp((-Vn - 35.0f) * (1.0f / 9.0f)));
      float aH = 0.25f  * fexp((-Vn - 90.0f) * (1.0f / 12.0f));
      float bN = -0.002f * (Vn - 25.0f) * frcp(1.0f - fexp((Vn - 25.0f) * (1.0f / 9.0f)));
      float bM = -0.124f * (Vn + 35.0f) * frcp(1.0f - fexp((Vn + 35.0f) * (1.0f / 9.0f)));
      float bH = 0.25f  * fexp((Vn + 34.0f) * (1.0f / 12.0f));

      // removable-singularity fixups (exact equality, as in the reference)
      bool at25  = (Vn == 25.0f);
      bool atm35 = (Vn == -35.0f);
      aN = at25  ? 0.18f  : aN;
      bN = at25  ? 0.08f  : bN;
      aM = atm35 ? 1.638f : aM;
      bM = atm35 ? 1.16f  : bM;

      // --- gating + synapse updates ---
      m = cn_update(aM, bM, m);
      n = cn_update(aN, bN, n);
      h = cn_update(aH, bH, h);
      float sy = 0.025f * (zp + 0.1f);          // DT/2*(A_D*z + A_R), A_D=1
      y = (zp * 0.05f + (1.0f - sy) * y) * frcp(sy + 1.0f);
      V = Vn;

      // --- output: sigmoid((V - VT)/KP) ---
      obase[(size_t)t * L_ + lane] =
          frcp(1.0f + fexp(-(Vn + 20.0f) * (1.0f / 3.0f)));
      ++t;
    }
  }
}

extern "C" void kernel_launch(void* const* d_in, const int* in_sizes, int n_in,
                              void* d_out, int out_size, void* d_ws, size_t ws_size,
                              hipStream_t stream) {
  (void)in_sizes; (void)n_in; (void)out_size; (void)d_ws; (void)ws_size;
  const float* z   = (const float*)d_in[0];
  float*       out = (float*)d_out;
  // 32 batches x 4 L-slices = 128 single-wave blocks, 32 threads each
  hh_synaptic_kernel<<<dim3(B_ * (L_ / LCHUNK)), dim3(LCHUNK), 0, stream>>>(z, out);
}